// AggregationGPE_21397527069066
// MI455X (gfx1250) — compile-verified
//
#include <hip/hip_runtime.h>
#include <hip/hip_bf16.h>

typedef float v2f __attribute__((ext_vector_type(2)));
typedef float v8f __attribute__((ext_vector_type(8)));

#define NFEAT   128
#define OUTDIM  384
#define PI_F    3.14159265358979323846f
#define TWO_PI  6.28318530717958647692f

__device__ __forceinline__ float fast_tanh(float t) {
#if __has_builtin(__builtin_amdgcn_tanh_f32)
    return __builtin_amdgcn_tanh_f32(t);
#elif __has_builtin(__builtin_amdgcn_tanhf)
    return __builtin_amdgcn_tanhf(t);
#else
    // tanh(t) = 1 - 2/(1 + exp(2t));  exp(2t) = exp2(t * 2*log2(e))
#if __has_builtin(__builtin_amdgcn_exp2f)
    float e = __builtin_amdgcn_exp2f(t * 2.88539008177792681472f);
#else
    float e = exp2f(t * 2.88539008177792681472f);
#endif
    return 1.0f - 2.0f * __builtin_amdgcn_rcpf(1.0f + e);
#endif
}

// One wave handles a 16-point x 16-feature tile of the output.
// Work decomposition: work = pt*24 + d*8 + ft
//   pt : point tile (16 consecutive flattened points of B*N)
//   d  : coordinate dim (0..2) -> output columns d*128 .. d*128+127
//   ft : feature sub-tile (0..7) -> 16 features
__global__ __launch_bounds__(256)
void AggregationGPE_kernel(const float* __restrict__ xyz,
                           float* __restrict__ out,
                           int n_point_tiles) {
    const int lane = threadIdx.x & 31;
    const int wave = threadIdx.x >> 5;
    const int work = blockIdx.x * 8 + wave;

    const int pt  = work / 24;
    if (pt >= n_point_tiles) return;          // wave-uniform: EXEC stays all-1s
    const int rem = work - pt * 24;
    const int d   = rem >> 3;                 // 0..2
    const int ft  = rem & 7;                  // 0..7

    const int  m0   = pt * 16;                // first flattened point of tile
    const int  l15  = lane & 15;
    const bool lo   = lane < 16;

    // x component (dim d) for point m0 + (lane&15); lanes 16-31 duplicate
    const float xv = xyz[(m0 + l15) * 3 + d];

    // per-point trig (shared across all 128 features of this dim)
    const float cx = __cosf(xv);
    const float sx = __sinf(xv);

    // per-feature basis constants; f = ft*16 + (lane&15), linspace index f+1
    const float fp = (float)(ft * 16 + l15 + 1);
    const float tv = fmaf(fp, TWO_PI / 129.0f, -PI_F);    // trig_vals[f]
    const float cv = __cosf(tv);
    const float sv = __sinf(tv);

    // A (16x4 f32): lanes 0-15 hold K=0,1 ; lanes 16-31 hold K=2,3 (zero pad)
    // B (4x16 f32): same K striping across lane halves
    v2f A, Bm;
    A.x  = lo ? cx : 0.0f;
    A.y  = lo ? sx : 0.0f;
    Bm.x = lo ? (cv - sv) : 0.0f;   // coeff of cos(x):  cos v - sin v
    Bm.y = lo ? (sv + cv) : 0.0f;   // coeff of sin(x):  sin v + cos v

    v8f C = {};
    // C[m][n] = cos(x_m - v_n) + sin(x_m - v_n)
    C = __builtin_amdgcn_wmma_f32_16x16x4_f32(
            /*neg_a=*/false, A, /*neg_b=*/false, Bm,
            /*c_mod=*/(short)0, C, /*reuse_a=*/false, /*reuse_b=*/false);

    // gaussian / tanh basis centers for this lane's feature column
    const float gv = fmaf(fp, 2.0f / 129.0f, -1.0f);  // gpe_vals[f]
    const float hv = fmaf(fp, 4.0f / 129.0f, -2.0f);  // tanh_vals[f]

    const int col   = d * NFEAT + (ft * 16 + l15);
    const int rowHi = (lane >> 4) << 3;               // 0 for lanes<16, 8 else

#pragma unroll
    for (int g = 0; g < 8; ++g) {
        // x of the point owning C-element (row = g + rowHi, col = lane&15)
        const float xg = __shfl(xv, g + rowHi, 32);

        const float gz    = xg - gv;
        const float gauss = __expf(-50.0f * gz * gz); // 0.5/sigma^2 = 50
        const float th    = fast_tanh(xg - hv);

        // softmax of 4 identical logits == 0.25 each
        const float val = 0.25f * (C[g] + gauss + th);

        const int row = m0 + g + rowHi;
        out[row * OUTDIM + col] = val;
    }
}

extern "C" void kernel_launch(void* const* d_in, const int* in_sizes, int n_in,
                              void* d_out, int out_size, void* d_ws, size_t ws_size,
                              hipStream_t stream) {
    const float* xyz = (const float*)d_in[0];
    // d_in[1] (neighbor_xyz) is provably unused: the curvature-derived softmax
    // weights are softmax of 4 equal logits == 0.25 regardless of curvature.
    float* out = (float*)d_out;

    const int n_points      = in_sizes[0] / 3;   // B*N flattened points
    const int n_point_tiles = n_points / 16;     // 16 points per tile
    const int works         = n_point_tiles * 24;
    const int blocks        = (works + 7) / 8;   // 8 waves (256 thr) per block

    AggregationGPE_kernel<<<blocks, 256, 0, stream>>>(xyz, out, n_point_tiles);
}